// GraphAttentionLayer_59416577573499
// MI455X (gfx1250) — compile-verified
//
#include <hip/hip_runtime.h>
#include <hip/hip_bf16.h>

// CDNA5 / gfx1250. wave32.
typedef __attribute__((ext_vector_type(2))) float v2f;
typedef __attribute__((ext_vector_type(8))) float v8f;

// ---------------- helpers ----------------

__device__ __forceinline__ float waveReduceSum(float v) {
    // full 32-lane butterfly reduction (wave32)
    #pragma unroll
    for (int m = 16; m >= 1; m >>= 1) v += __shfl_xor(v, m, 32);
    return v;
}

__device__ __forceinline__ float gumbel_hash(unsigned idx) {
    // deterministic stand-in for jax.random gumbel noise (threefry not reproduced)
    unsigned x = idx * 0x9E3779B1u + 0x85EBCA6Bu;
    x ^= x >> 16; x *= 0x7FEB352Du;
    x ^= x >> 15; x *= 0x846CA68Bu;
    x ^= x >> 16;
    float u = (float)(x >> 8) * (1.0f / 16777216.0f);
    u = fmaxf(u, 1e-20f);
    return -logf(-logf(u));
}

// ---------------- kernel 1: one chain step  cum_out = alpha[k]*(cum_in @ rel) + node[k+1] ----------------
// rel[b,p] = sum_r relw[b,p,k,r] * W[r]  => acc = sum_r w_r * (cum_in @ W_r)  via f32 WMMA.
// 1 wave = 16 paths (M tile). A frags (16x4 f32 layout) reused across 9 r * 4 e-tiles.
// W[r] staged transposed in LDS so B frags (4x16) are contiguous b64 LDS loads.

__global__ __launch_bounds__(256) void k1_chain(
    const float* __restrict__ cumIn, int inStride,
    const float* __restrict__ node,   // path_node_embed (8000, 3, 64)
    const float* __restrict__ relw,   // path_relation  (8000, 3, 9)
    const float* __restrict__ W,      // (9, 64, 64)
    const float* __restrict__ alpha,  // (3)
    int k,
    float* __restrict__ cumOut)       // (8000, 64)
{
    __shared__ float wt[64 * 64];     // wt[e*64 + d] = W[r, d, e]  (16 KB)

    const int tid  = threadIdx.x;
    const int wave = tid >> 5;
    const int lane = tid & 31;
    const int tile = blockIdx.x * 8 + wave;    // 500 tiles of 16 rows
    const bool active = (tile < 500);
    const int n16 = lane & 15;
    const int hi  = lane >> 4;                 // 0: K/M group low, 1: high
    const int pbase = tile * 16;
    const float alph = alpha[k];

    // A fragments: 16x4 f32 layout. lane<16 holds (K=4kk+0,1), lane>=16 holds (K=4kk+2,3), row = pbase+n16.
    v2f afr[16];
    if (active) {
        const float* ap = cumIn + (size_t)(pbase + n16) * (size_t)inStride + 2 * hi;
        #pragma unroll
        for (int kk = 0; kk < 16; ++kk)
            afr[kk] = *(const v2f*)(ap + 4 * kk);
    }

    v8f acc[4] = {};   // 4 e-tiles of 16 cols, C/D 16x16 layout

    for (int r = 0; r < 9; ++r) {
        __syncthreads();                       // LDS reuse fence
        // stage transpose of W[r]
        const float* Wr = W + r * 4096;
        for (int i = tid; i < 4096; i += 256) {
            int d = i >> 6, e = i & 63;
            wt[e * 64 + d] = Wr[i];
        }
        __syncthreads();
        if (!active) continue;

        // per-output-row relation weights for this r (row m = j + 8*hi in C/D layout)
        float wrow[8];
        #pragma unroll
        for (int j = 0; j < 8; ++j) {
            int row = pbase + j + 8 * hi;
            wrow[j] = relw[(row * 3 + k) * 9 + r];
        }

        #pragma unroll
        for (int et = 0; et < 4; ++et) {
            v8f c = {};
            const int e = et * 16 + n16;
            const float* bp = &wt[e * 64 + 2 * hi];   // B frag: rows 4kk+2*hi, 4kk+2*hi+1 ; col e
            #pragma unroll
            for (int kk = 0; kk < 16; ++kk) {
                v2f b = *(const v2f*)(bp + 4 * kk);
                c = __builtin_amdgcn_wmma_f32_16x16x4_f32(
                        /*neg_a=*/false, afr[kk],
                        /*neg_b=*/false, b,
                        /*c_mod=*/(short)0, c,
                        /*reuse_a=*/false, /*reuse_b=*/false);
            }
            #pragma unroll
            for (int j = 0; j < 8; ++j) acc[et][j] += wrow[j] * c[j];
        }
    }

    if (active) {
        #pragma unroll
        for (int et = 0; et < 4; ++et) {
            const int e = et * 16 + n16;
            #pragma unroll
            for (int j = 0; j < 8; ++j) {
                int row = pbase + j + 8 * hi;
                float v = alph * acc[et][j];
                if (k < 2) v += node[(size_t)row * 192 + (k + 1) * 64 + e];
                cumOut[(size_t)row * 64 + e] = v;
            }
        }
    }
}

// ---------------- kernel 2: per (b,v,f) gate + path attention + M_j + causal sigmoid ----------------
// 800 groups, one wave each. Lane owns d-pair (2*lane, 2*lane+1) of hd=64.

__global__ __launch_bounds__(256) void k2_attn(
    const float* __restrict__ cum,     // (8000, 64) final chain output
    const float* __restrict__ h_time,  // (4, 64)
    const float* __restrict__ Wp,      // (128)
    const float* __restrict__ cw,      // (128) causal_w[0]
    const float* __restrict__ cb,      // (1)
    float* __restrict__ mj,            // ws (800, 64)
    float* __restrict__ out)           // packed outputs
{
    const int w = blockIdx.x * 8 + (threadIdx.x >> 5);
    const int lane = threadIdx.x & 31;
    if (w >= 800) return;
    const int b  = w / 200;
    const int rp = w * 10;                 // first path row of this (b,v,f)
    const int d0 = lane * 2, d1 = d0 + 1;

    const float ht0 = h_time[b * 64 + d0];
    const float ht1 = h_time[b * 64 + d1];
    const float hdot = waveReduceSum(ht0 * Wp[d0] + ht1 * Wp[d1]);

    // sigmoid logits + gumbel, hard one-hot via argmax (forward value of ST-gumbel)
    float x[10];
    #pragma unroll
    for (int p = 0; p < 10; ++p) {
        const float* cr = cum + (size_t)(rp + p) * 64;
        float s = waveReduceSum(cr[d0] * Wp[64 + d0] + cr[d1] * Wp[64 + d1]);
        float logit = 1.0f / (1.0f + expf(-(hdot + s)));
        x[p] = logit + gumbel_hash((unsigned)(rp + p));
    }
    int amax = 0;
    #pragma unroll
    for (int p = 1; p < 10; ++p) if (x[p] > x[amax]) amax = p;

    // path_attn = softmax(one_hot with zeros replaced by 1e-6)
    float e[10], sum = 0.0f;
    #pragma unroll
    for (int p = 0; p < 10; ++p) { e[p] = expf(p == amax ? 1.0f : 1e-6f); sum += e[p]; }

    float mj0 = 0.0f, mj1 = 0.0f;
    #pragma unroll
    for (int p = 0; p < 10; ++p) {
        float a = e[p] / sum;
        if (lane == 0) out[7680 + rp + p] = a;            // path_attn
        const float* cr = cum + (size_t)(rp + p) * 64;
        mj0 += a * cr[d0];
        mj1 += a * cr[d1];
    }

    // joint_impact: only diag(cs) survives einsum('bvff,bvfd->bvfd')
    float norm2 = waveReduceSum(mj0 * mj0 + mj1 * mj1);
    float den = fmaxf(sqrtf(norm2), 1e-12f);
    float cs = norm2 / (den * den);
    mj0 *= cs; mj1 *= cs;

    // causal sigmoid gate
    float s = waveReduceSum(ht0 * cw[d0] + ht1 * cw[d1] + mj0 * cw[64 + d0] + mj1 * cw[64 + d1]) + cb[0];
    float ac = 1.0f / (1.0f + expf(-s));

    mj[w * 64 + d0] = mj0;
    mj[w * 64 + d1] = mj1;
    if (lane == 0) out[15680 + w] = ac;                    // attn_causal
}

// ---------------- kernel 3: g_c / g_t reductions + II gather for g_i ----------------

__global__ __launch_bounds__(256) void k3_reduce(
    const float* __restrict__ mj,      // (800, 64)
    const float* __restrict__ acbuf,   // attn_causal (800), lives in out[15680..]
    const int*   __restrict__ II,      // (4)
    float* __restrict__ out)
{
    const int t = blockIdx.x * blockDim.x + threadIdx.x;
    if (t >= 2560) return;
    const int d  = t & 63;
    const int bv = t >> 6;             // (b*10+v), 0..39
    const int b  = bv / 10, v = bv % 10;

    float gc = 0.0f, gt = 0.0f;
    for (int f = 0; f < 20; ++f) {
        int g = bv * 20 + f;
        float m = mj[g * 64 + d];
        float a = acbuf[g];
        gc += a * m;
        gt += (1.0f - a) * m;
    }
    int b2 = II[b];
    int bv2 = b2 * 10 + v;
    float gt2 = 0.0f;
    for (int f = 0; f < 20; ++f) {
        int g = bv2 * 20 + f;
        gt2 += (1.0f - acbuf[g]) * mj[g * 64 + d];
    }
    out[t]        = gc + gt2;   // g_i
    out[2560 + t] = gc;         // g_c
    out[5120 + t] = gt;         // g_t
}

// ---------------- launch ----------------
// inputs: 0 h_time, 1 path_index(unused), 2 path_node_embed, 3 path_relation, 4 W,
//         5 alpha, 6 W_p, 7 causal_w, 8 causal_b, 9 II, 10..13 flags (fixed 1,1,1,0)

extern "C" void kernel_launch(void* const* d_in, const int* in_sizes, int n_in,
                              void* d_out, int out_size, void* d_ws, size_t ws_size,
                              hipStream_t stream) {
    const float* h_time = (const float*)d_in[0];
    const float* node   = (const float*)d_in[2];
    const float* relw   = (const float*)d_in[3];
    const float* W      = (const float*)d_in[4];
    const float* alpha  = (const float*)d_in[5];
    const float* Wp     = (const float*)d_in[6];
    const float* cw     = (const float*)d_in[7];
    const float* cb     = (const float*)d_in[8];
    const int*   II     = (const int*)d_in[9];
    float* out = (float*)d_out;

    char* ws = (char*)d_ws;
    float* cumA = (float*)ws;                                   // 8000*64 f32
    float* cumB = (float*)(ws + 8000 * 64 * sizeof(float));     // 8000*64 f32
    float* mj   = (float*)(ws + 2 * 8000 * 64 * sizeof(float)); // 800*64 f32

    // K=3 sequential chain; k=0 reads node[...,0,:] directly (row stride 192 floats)
    k1_chain<<<63, 256, 0, stream>>>(node, 192, node, relw, W, alpha, 0, cumA);
    k1_chain<<<63, 256, 0, stream>>>(cumA,  64, node, relw, W, alpha, 1, cumB);
    k1_chain<<<63, 256, 0, stream>>>(cumB,  64, node, relw, W, alpha, 2, cumA);

    k2_attn<<<100, 256, 0, stream>>>(cumA, h_time, Wp, cw, cb, mj, out);
    k3_reduce<<<10, 256, 0, stream>>>(mj, out + 15680, II, out);
}